// GravNetLayer_9663676416361
// MI455X (gfx1250) — compile-verified
//
#include <hip/hip_runtime.h>
#include <hip/hip_bf16.h>
#include <math.h>

// ---------------------------------------------------------------------------
// GravNet layer for MI455X (gfx1250, wave32, WMMA).
// fp32 WMMA (V_WMMA_F32_16X16X4_F32): the op is ~1.6 GFLOP / ~50 MB traffic
// (~2us at 23.3 TB/s HBM) -> latency bound; full fp32 keeps KNN selection
// faithful to the reference at zero cost.
//
// Round-1 fix: all wave-uniform selects hoisted out of the WMMA inner loops,
// B fragments preloaded into registers once per wave and reused across RT=4
// row tiles, so the steady-state loop is global_load_b64 + v_wmma only.
// ---------------------------------------------------------------------------

typedef float v2f __attribute__((ext_vector_type(2)));
typedef float v8f __attribute__((ext_vector_type(8)));

__device__ __forceinline__ v8f wmma_f32_k4(v2f a, v2f b, v8f c) {
  // (neg_a, A, neg_b, B, c_mod, C, reuse_a, reuse_b)
  return __builtin_amdgcn_wmma_f32_16x16x4_f32(false, a, false, b, (short)0, c,
                                               false, false);
}

namespace cfg {
constexpr int B    = 8;
constexpr int N    = 2048;
constexpr int Din  = 128;
constexpr int Ds   = 4;
constexpr int Dp   = 64;
constexpr int Dout = 128;
constexpr int KNN  = 16;
constexpr int BN   = B * N;
constexpr int RT   = 4;   // row tiles per wave (B-fragment reuse factor)
constexpr int KS   = 32;  // k-steps of 4 covering K=128
}  // namespace cfg

// ---------------------------------------------------------------------------
// Kernel 1: coords = x@W_space + b_space ; feats = (x@W_feat + b_feat)*mask
// One wave per (RT*16-row super tile, 16-col tile). blockIdx.y in [0,5):
//   0..3 -> feats column tiles, 4 -> coords tile (only 4 valid columns).
// A fragment (16x4 f32): lane L holds row (L&15), k = 4s + 2*(L>>4) + {0,1}.
// B fragment (4x16 f32): lane L holds col (L&15), same k pair.
// Coords tile: B columns 4..15 are garbage-by-construction (clamped loads) --
// harmless because C column c depends only on B column c and we never store
// columns >= 4.
// ---------------------------------------------------------------------------
__global__ __launch_bounds__(32) void gravnet_encode_kernel(
    const float* __restrict__ x, const unsigned char* __restrict__ mask,
    const float* __restrict__ Wsp, const float* __restrict__ bsp,
    const float* __restrict__ Wft, const float* __restrict__ bft,
    float* __restrict__ coords, float* __restrict__ feats) {
  using namespace cfg;
  const int lane = threadIdx.x;
  const int g    = blockIdx.x;  // super row tile (RT*16 rows)
  const int t    = blockIdx.y;  // 0..3 feats tiles, 4 = coords
  const int m    = lane & 15;
  const int koff = (lane >> 4) * 2;
  const int ncol = lane & 15;

  // -------- preload all B fragments into registers (64 VGPRs) --------
  v2f bf[KS];
  if (t < 4) {
    const int col = t * 16 + ncol;
#pragma unroll
    for (int s = 0; s < KS; ++s) {
      const int k = s * 4 + koff;
      bf[s].x = Wft[(size_t)k * Dp + col];
      bf[s].y = Wft[(size_t)(k + 1) * Dp + col];
    }
  } else {
    const int colc = (ncol < Ds) ? ncol : (Ds - 1);  // clamp: stay in bounds
#pragma unroll
    for (int s = 0; s < KS; ++s) {
      const int k = s * 4 + koff;
      bf[s].x = Wsp[(size_t)k * Ds + colc];
      bf[s].y = Wsp[(size_t)(k + 1) * Ds + colc];
    }
  }

  for (int rt = 0; rt < RT; ++rt) {
    const int rtile   = g * RT + rt;
    const size_t rowA = (size_t)rtile * 16 + m;
    const float* xrow = x + rowA * Din + koff;

    v8f acc = {0.f, 0.f, 0.f, 0.f, 0.f, 0.f, 0.f, 0.f};
#pragma unroll
    for (int s = 0; s < KS; ++s) {
      const v2f a = *(const v2f*)(xrow + s * 4);
      acc = wmma_f32_k4(a, bf[s], acc);
    }

    const int rhalf = (lane >> 4) * 8;  // C/D: row = v + 8*(lane>>4)
    if (t < 4) {
      const int col    = t * 16 + ncol;
      const float bias = bft[col];
#pragma unroll
      for (int v = 0; v < 8; ++v) {
        const size_t row = (size_t)rtile * 16 + rhalf + v;
        const float mf   = mask[row] ? 1.0f : 0.0f;
        feats[row * Dp + col] = (acc[v] + bias) * mf;
      }
    } else if (ncol < Ds) {
      const float bias = bsp[ncol];
#pragma unroll
      for (int v = 0; v < 8; ++v) {
        const size_t row = (size_t)rtile * 16 + rhalf + v;
        coords[row * Ds + ncol] = acc[v] + bias;
      }
    }
  }
}

// ---------------------------------------------------------------------------
// Kernel 2: per-batch KNN (top-16 by squared distance in 4-D coord space)
// + Gaussian-weighted neighbor-feature mean.
// 512 threads = 16 waves per block; one query per wave; batch coords (masked:
// invalid -> +inf, matching coords_masked in the reference) staged in 32KB LDS.
// Per-lane sorted top-16 in registers (unrolled compare-swap insertion),
// merged across the wave with 16 rounds of shuffle argmin.
// ---------------------------------------------------------------------------
__global__ __launch_bounds__(512) void gravnet_knn_kernel(
    const float* __restrict__ coords, const unsigned char* __restrict__ mask,
    const float* __restrict__ feats, float* __restrict__ wmean) {
  using namespace cfg;
  __shared__ float4 sc[N];  // 32 KB of the 320 KB WGP LDS

  const int b      = blockIdx.y;
  const int tid    = threadIdx.x;
  const float FINF = __builtin_inff();

  for (int j = tid; j < N; j += 512) {
    const float* cp = coords + ((size_t)b * N + j) * Ds;
    float4 c4 = *(const float4*)cp;
    if (!mask[b * N + j]) c4 = make_float4(FINF, FINF, FINF, FINF);
    sc[j] = c4;
  }
  __syncthreads();

  const int wave = tid >> 5;
  const int lane = tid & 31;
  const int q    = blockIdx.x * 16 + wave;

  const float* qp = coords + ((size_t)b * N + q) * Ds;  // unmasked query coords
  const float qx = qp[0], qy = qp[1], qz = qp[2], qw = qp[3];

  float bd[KNN];
  int   bi[KNN];
#pragma unroll
  for (int i = 0; i < KNN; ++i) { bd[i] = FINF; bi[i] = -1; }

  for (int j = lane; j < N; j += 32) {
    const float4 c4 = sc[j];
    const float dx = qx - c4.x, dy = qy - c4.y, dz = qz - c4.z, dw = qw - c4.w;
    float d2 = dx * dx + dy * dy + dz * dz + dw * dw;
    if (d2 < bd[KNN - 1]) {
      int jj = j;
#pragma unroll
      for (int i = 0; i < KNN; ++i) {
        if (d2 < bd[i]) {
          const float td = bd[i]; bd[i] = d2; d2 = td;
          const int   ti = bi[i]; bi[i] = jj; jj = ti;
        }
      }
    }
  }

  // Merge sorted per-lane lists: 16 rounds of wave32 argmin on list heads.
  float topd[KNN];
  int   topi[KNN];
#pragma unroll
  for (int r = 0; r < KNN; ++r) {
    float v = bd[0];
    int   l = lane;
#pragma unroll
    for (int off = 16; off >= 1; off >>= 1) {
      const float ov = __shfl_xor(v, off);
      const int   ol = __shfl_xor(l, off);
      if (ov < v || (ov == v && ol < l)) { v = ov; l = ol; }
    }
    topd[r] = v;
    topi[r] = __shfl(bi[0], l);
    if (lane == l) {  // winner pops its head
#pragma unroll
      for (int i = 0; i < KNN - 1; ++i) { bd[i] = bd[i + 1]; bi[i] = bi[i + 1]; }
      bd[KNN - 1] = FINF;
      bi[KNN - 1] = -1;
    }
  }

  // weights = exp(-10*d2); weighted mean of neighbor features (Dp=64 -> 2/lane)
  float w[KNN];
  float wsum = 0.0f;
#pragma unroll
  for (int r = 0; r < KNN; ++r) {
    const float wr = (topi[r] >= 0) ? __expf(-10.0f * topd[r]) : 0.0f;
    w[r] = wr;
    wsum += wr;
  }
  const float inv = 1.0f / fmaxf(wsum, 1e-8f);

  float acc0 = 0.0f, acc1 = 0.0f;
#pragma unroll
  for (int r = 0; r < KNN; ++r) {
    const int idx = topi[r];  // wave-uniform
    if (idx >= 0) {
      const float* fp = feats + ((size_t)b * N + idx) * Dp;
      acc0 += w[r] * fp[lane];
      acc1 += w[r] * fp[lane + 32];
    }
  }
  float* op = wmean + ((size_t)b * N + q) * Dp;
  op[lane]      = acc0 * inv;
  op[lane + 32] = acc1 * inv;
}

// ---------------------------------------------------------------------------
// Kernels 3/4: out = act([lo | hi] @ W + bias). K=128 split into a lo-half and
// hi-half loop (handles the concat in layer 1; layer 2 uses lo=h, hi=h+64).
// B fragments register-resident, reused across RT row tiles per wave.
// ---------------------------------------------------------------------------
__global__ __launch_bounds__(32) void gravnet_mlp_kernel(
    const float* __restrict__ lo, const float* __restrict__ hi, int strideIn,
    const float* __restrict__ W, const float* __restrict__ bias,
    float* __restrict__ out, int relu) {
  using namespace cfg;
  const int lane = threadIdx.x;
  const int g    = blockIdx.x;
  const int t    = blockIdx.y;  // 0..7 output column tiles
  const int m    = lane & 15;
  const int koff = (lane >> 4) * 2;
  const int ncol = lane & 15;
  const int col  = t * 16 + ncol;

  v2f bf[KS];
#pragma unroll
  for (int s = 0; s < KS; ++s) {
    const int k = s * 4 + koff;
    bf[s].x = W[(size_t)k * Dout + col];
    bf[s].y = W[(size_t)(k + 1) * Dout + col];
  }
  const float bv = bias[col];

  for (int rt = 0; rt < RT; ++rt) {
    const int rtile   = g * RT + rt;
    const size_t rowA = (size_t)rtile * 16 + m;
    const float* plo  = lo + rowA * strideIn + koff;
    const float* phi  = hi + rowA * strideIn + koff;

    v8f acc = {0.f, 0.f, 0.f, 0.f, 0.f, 0.f, 0.f, 0.f};
#pragma unroll
    for (int s = 0; s < KS / 2; ++s) {  // k in [0,64): lo half
      const v2f a = *(const v2f*)(plo + s * 4);
      acc = wmma_f32_k4(a, bf[s], acc);
    }
#pragma unroll
    for (int s = KS / 2; s < KS; ++s) {  // k in [64,128): hi half
      const v2f a = *(const v2f*)(phi + (s - KS / 2) * 4);
      acc = wmma_f32_k4(a, bf[s], acc);
    }

    const int rhalf = (lane >> 4) * 8;
#pragma unroll
    for (int v = 0; v < 8; ++v) {
      const size_t row = (size_t)rtile * 16 + rhalf + v;
      float val = acc[v] + bv;
      if (relu) val = fmaxf(val, 0.0f);
      out[row * Dout + col] = val;
    }
  }
}

// ---------------------------------------------------------------------------
// Launch. Inputs (setup_inputs order):
//  0 x[8,2048,128] f32, 1 mask[8,2048] bool(u8), 2 W_space[128,4], 3 b_space[4],
//  4 W_feat[128,64], 5 b_feat[64], 6 W1[128,128], 7 b1[128], 8 W2[128,128],
//  9 b2[128]. Output: [8,2048,128] f32.
// Workspace layout (floats): coords BN*4 | feats BN*64 | wmean BN*64 | h BN*128
// = ~17 MB.
// ---------------------------------------------------------------------------
extern "C" void kernel_launch(void* const* d_in, const int* in_sizes, int n_in,
                              void* d_out, int out_size, void* d_ws,
                              size_t ws_size, hipStream_t stream) {
  using namespace cfg;
  const float* x          = (const float*)d_in[0];
  const unsigned char* mk = (const unsigned char*)d_in[1];
  const float* Wsp        = (const float*)d_in[2];
  const float* bsp        = (const float*)d_in[3];
  const float* Wft        = (const float*)d_in[4];
  const float* bft        = (const float*)d_in[5];
  const float* W1         = (const float*)d_in[6];
  const float* b1         = (const float*)d_in[7];
  const float* W2         = (const float*)d_in[8];
  const float* b2         = (const float*)d_in[9];
  float* out              = (float*)d_out;

  float* coords = (float*)d_ws;
  float* feats  = coords + (size_t)BN * Ds;
  float* wmean  = feats + (size_t)BN * Dp;
  float* h      = wmean + (size_t)BN * Dp;

  gravnet_encode_kernel<<<dim3(BN / (16 * RT), 5), 32, 0, stream>>>(
      x, mk, Wsp, bsp, Wft, bft, coords, feats);
  gravnet_knn_kernel<<<dim3(N / 16, B), 512, 0, stream>>>(coords, mk, feats,
                                                          wmean);
  gravnet_mlp_kernel<<<dim3(BN / (16 * RT), Dout / 16), 32, 0, stream>>>(
      feats, wmean, Dp, W1, b1, h, 1);
  gravnet_mlp_kernel<<<dim3(BN / (16 * RT), Dout / 16), 32, 0, stream>>>(
      h, h + 64, Dout, W2, b2, out, 0);
}